// PointNetFeaturePropagation_46712064311940
// MI455X (gfx1250) — compile-verified
//
#include <hip/hip_runtime.h>

typedef __attribute__((ext_vector_type(2))) float v2f;
typedef __attribute__((ext_vector_type(4))) float v4f;
typedef __attribute__((ext_vector_type(8))) float v8f;
typedef __attribute__((ext_vector_type(4))) unsigned int u32x4;
typedef __attribute__((ext_vector_type(4))) int i32x4;
typedef __attribute__((ext_vector_type(8))) int i32x8;

namespace {
constexpr int BB   = 8;
constexpr int NN   = 8192;
constexpr int SS   = 2048;
constexpr int D1   = 128;
constexpr int D2   = 256;
constexpr int CIN  = D1 + D2;   // 384
constexpr int CMID = 256;
constexpr int COUT = 256;
constexpr int NCOL = BB * NN;   // 65536 GEMM columns

constexpr int KCH     = 64;     // K chunk per TDM stage
constexpr int ASTRIDE = 66;     // padded LDS row stride for A panel (bank spread, b64-aligned)
constexpr int BSTRIDE = 80;     // padded LDS row stride for B tile (TDM pad: 64 + 16 dwords)
}

// ---------------------------------------------------------------------------
// Tensor Data Mover: stage a dense 2D f32 tile (tile_d1 rows x tile_d0 elems,
// row stride row_stride elems) from global into LDS at lds_off, adding a
// 16-DWORD pad after every 64 DWORDs (=> LDS row stride 80 dwords).
// Descriptor packing per CDNA5 ISA. Issued per-wave
// (EXEC ignored) -> caller must gate to a single wave. Tracked by TENSORcnt.
// ---------------------------------------------------------------------------
__device__ __forceinline__ void tdm_load_tile_2d(
    unsigned lds_off, const void* gptr,
    unsigned tensor_d0, unsigned tensor_d1,
    unsigned tile_d0, unsigned tile_d1,
    unsigned long long row_stride)
{
  unsigned long long ga = (unsigned long long)(uintptr_t)gptr;
  u32x4 g0;
  g0[0] = 1u;                                    // count=1 (valid), user mode
  g0[1] = lds_off;                               // lds_addr (bytes)
  g0[2] = (unsigned)ga;                          // global_addr[31:0]  (bits 95:64)
  g0[3] = (unsigned)((ga >> 32) & 0x01ffffffu)   // global_addr[56:32] (bits 120:96)
          | 0x80000000u;                         // type=2 ("image")   (bits 127:126)
  i32x8 g1;
  g1[0] = (int)((2u << 16)                       // data_size = 4 bytes
              | (1u << 20)                       // pad_enable
              | (5u << 22)                       // pad_interval: 64 DWORDs
              | (15u << 25));                    // pad_amount: 16 DWORDs
  g1[1] = (int)((tensor_d0 & 0xffffu) << 16);                       // tensor_dim0[15:0]
  g1[2] = (int)((tensor_d0 >> 16) | ((tensor_d1 & 0xffffu) << 16)); // dim0[31:16]|dim1[15:0]
  g1[3] = (int)((tensor_d1 >> 16) | (tile_d0 << 16));               // dim1[31:16]|tile_dim0
  g1[4] = (int)(tile_d1 & 0xffffu);                                 // tile_dim1, tile_dim2=0
  g1[5] = (int)(unsigned)(row_stride & 0xffffffffull);              // tensor_dim0_stride[31:0]
  g1[6] = (int)(unsigned)((row_stride >> 32) & 0xffffull);          // stride[47:32]
  g1[7] = 0;
  i32x4 z4 = {0, 0, 0, 0};
#if __clang_major__ >= 23
  i32x8 z8 = {0, 0, 0, 0, 0, 0, 0, 0};
  __builtin_amdgcn_tensor_load_to_lds(g0, g1, z4, z4, z8, 0);
#else
  __builtin_amdgcn_tensor_load_to_lds(g0, g1, z4, z4, 0);
#endif
}

// ---------------------------------------------------------------------------
// Kernel 1: fused sq-dist + top-3 + interp + concat.
// Block = 256 threads handles 64 query points: 4 threads per point scan
// disjoint strided quarters of S (s = 4*it + g -> disjoint LDS bank groups),
// per-segment top-3 in registers, then a 12-candidate lexicographic merge
// per point (reproduces top_k lowest-index-on-ties), then a flat
// (channel x point) interpolation/concat phase with coalesced stores.
// ---------------------------------------------------------------------------
__global__ __launch_bounds__(256) void k_topk_interp(
    const float* __restrict__ xyz1, const float* __restrict__ xyz2,
    const float* __restrict__ points1, const float* __restrict__ points2,
    float* __restrict__ NP)
{
  __shared__ v4f   ref[SS];          // 32 KB: {x, y, z, |p|^2}
  __shared__ float cd[256 * 3];      // per-(point,segment) candidate dists
  __shared__ int   ci[256 * 3];      // per-(point,segment) candidate idx
  __shared__ float fw[64 * 3];       // final weights per point
  __shared__ int   fi[64 * 3];       // final indices per point
  const int tid = threadIdx.x;
  const int b   = blockIdx.x >> 7;             // N/64 = 128 chunks per batch
  const int n0  = (blockIdx.x & 127) << 6;

  const float* x2 = xyz2 + (size_t)b * 3 * SS;
  for (int s = tid; s < SS; s += 256) {
    float xx = x2[s], yy = x2[SS + s], zz = x2[2 * SS + s];
    v4f r; r.x = xx; r.y = yy; r.z = zz; r.w = xx * xx + yy * yy + zz * zz;
    ref[s] = r;
  }
  __syncthreads();

  // ---- phase 1: 4-way split scan ----
  const int p = tid >> 2, g = tid & 3;
  const int n = n0 + p;
  const float px = xyz1[(size_t)b * 3 * NN + n];
  const float py = xyz1[(size_t)b * 3 * NN + NN + n];
  const float pz = xyz1[(size_t)b * 3 * NN + 2 * NN + n];
  const float q1 = px * px + py * py + pz * pz;

  float d0 = 3.4e38f, d1 = 3.4e38f, d2 = 3.4e38f;
  int   i0 = -1, i1 = -1, i2 = -1;
  for (int it = 0; it < SS / 4; ++it) {
    const int s = (it << 2) | g;                 // increasing within segment
    v4f r = ref[s];
    float d = q1 + r.w - 2.0f * (px * r.x + py * r.y + pz * r.z);
    bool c0 = d < d0, c1 = d < d1, c2 = d < d2;  // strict < keeps earliest s
    float nd2 = c1 ? d1 : (c2 ? d : d2); int ni2 = c1 ? i1 : (c2 ? s : i2);
    float nd1 = c0 ? d0 : (c1 ? d : d1); int ni1 = c0 ? i0 : (c1 ? s : i1);
    float nd0 = c0 ? d  : d0;            int ni0 = c0 ? s  : i0;
    d0 = nd0; d1 = nd1; d2 = nd2; i0 = ni0; i1 = ni1; i2 = ni2;
  }
  cd[tid * 3 + 0] = d0; ci[tid * 3 + 0] = i0;
  cd[tid * 3 + 1] = d1; ci[tid * 3 + 1] = i1;
  cd[tid * 3 + 2] = d2; ci[tid * 3 + 2] = i2;
  __syncthreads();

  // ---- phase 2: merge 12 candidates per point (lexicographic (d, idx)) ----
  if (tid < 64) {
    float D0 = 3.4e38f, D1v = 3.4e38f, D2v = 3.4e38f;
    int   I0 = 0x7fffffff, I1 = 0x7fffffff, I2 = 0x7fffffff;
    for (int g2 = 0; g2 < 4; ++g2) {
      const int base = (tid * 4 + g2) * 3;
      for (int k = 0; k < 3; ++k) {
        float d = cd[base + k]; int i = ci[base + k];
        bool l0 = (d < D0)  || (d == D0  && i < I0);
        bool l1 = (d < D1v) || (d == D1v && i < I1);
        bool l2 = (d < D2v) || (d == D2v && i < I2);
        float nD2 = l1 ? D1v : (l2 ? d : D2v); int nI2 = l1 ? I1 : (l2 ? i : I2);
        float nD1 = l0 ? D0  : (l1 ? d : D1v); int nI1 = l0 ? I0 : (l1 ? i : I1);
        float nD0 = l0 ? d   : D0;             int nI0 = l0 ? i  : I0;
        D0 = nD0; D1v = nD1; D2v = nD2; I0 = nI0; I1 = nI1; I2 = nI2;
      }
    }
    float r0 = 1.0f / (D0  + 1e-8f);
    float r1 = 1.0f / (D1v + 1e-8f);
    float r2 = 1.0f / (D2v + 1e-8f);
    float rs = 1.0f / (r0 + r1 + r2);
    fw[tid * 3 + 0] = r0 * rs; fi[tid * 3 + 0] = I0;
    fw[tid * 3 + 1] = r1 * rs; fi[tid * 3 + 1] = I1;
    fw[tid * 3 + 2] = r2 * rs; fi[tid * 3 + 2] = I2;
  }
  __syncthreads();

  // ---- phase 3: interpolate + concat, (channel x point) parallel ----
  const float* p2 = points2 + (size_t)b * D2 * SS;
  const float* p1 = points1 + (size_t)b * D1 * NN;
  const int colbase = b * NN + n0;
  for (int e = tid; e < CIN * 64; e += 256) {
    const int c = e >> 6, j = e & 63;            // c uniform per wave
    const int col = colbase + j;
    if (c < D2) {
      const float w0 = fw[j * 3 + 0], w1 = fw[j * 3 + 1], w2 = fw[j * 3 + 2];
      const int   a0 = fi[j * 3 + 0], a1 = fi[j * 3 + 1], a2 = fi[j * 3 + 2];
      const float* row = p2 + (size_t)c * SS;
      NP[(size_t)c * NCOL + col] = w0 * row[a0] + w1 * row[a1] + w2 * row[a2];
    } else {
      NP[(size_t)c * NCOL + col] = p1[(size_t)(c - D2) * NN + n0 + j];
    }
  }
}

// ---------------------------------------------------------------------------
// fp32 WMMA (16x16x4) GEMM: block = 8 waves = 128 output rows x 64 cols.
// A panel staged in LDS (padded stride 66); B tile double-buffered in LDS by
// the Tensor Data Mover (padded stride 80) and overlapped with compute.
// ---------------------------------------------------------------------------
#define WMMA_F32(A_, B_, C_) \
  __builtin_amdgcn_wmma_f32_16x16x4_f32(false, (A_), false, (B_), (short)0, (C_), false, false)

__global__ __launch_bounds__(256) void k_conv0_wmma(
    const float* __restrict__ NP, const float* __restrict__ W0,
    const float* __restrict__ bias0, float* __restrict__ H)
{
  __shared__ float As[128 * ASTRIDE];          // ~33 KB
  __shared__ float Bs[2][KCH * BSTRIDE];       // 40 KB, TDM double buffer
  const int tid  = threadIdx.x;
  const int lane = tid & 31;
  const int wave = tid >> 5;
  const int lo = lane & 15, hi = lane >> 4;
  const int rowbase = blockIdx.y * 128;
  const int j0 = blockIdx.x * 64;

  v8f acc0 = {}, acc1 = {}, acc2 = {}, acc3 = {};

  constexpr int nchunk = CIN / KCH;            // 6
  if (wave == 0) {
    tdm_load_tile_2d((unsigned)(uintptr_t)&Bs[0][0], NP + j0,
                     NCOL, CIN, 64, KCH, NCOL);
  }

  for (int c = 0; c < nchunk; ++c) {
    const int kc = c * KCH;
    __syncthreads();   // all waves done reading Bs[(c+1)&1] and As from prev iter
    for (int i = tid; i < 128 * 32; i += 256) {
      int m = i >> 5, k2 = (i & 31) << 1;
      *(v2f*)(As + m * ASTRIDE + k2) =
          *(const v2f*)(W0 + (size_t)(rowbase + m) * CIN + kc + k2);
    }
    if (wave == 0) {
      if (c + 1 < nchunk) {
        tdm_load_tile_2d((unsigned)(uintptr_t)&Bs[(c + 1) & 1][0],
                         NP + (size_t)(kc + KCH) * NCOL + j0,
                         NCOL, (unsigned)(CIN - kc - KCH), 64, KCH, NCOL);
        __builtin_amdgcn_s_wait_tensorcnt(1);   // chunk c landed
      } else {
        __builtin_amdgcn_s_wait_tensorcnt(0);
      }
    }
    __syncthreads();   // As stored + Bs[c&1] visible to all waves

    const float* bsrc = Bs[c & 1];
    const float* arow = As + (wave * 16 + lo) * ASTRIDE + 2 * hi;
    for (int kk = 0; kk < KCH; kk += 4) {
      const v2f a = *(const v2f*)(arow + kk);
      const float* bp0 = bsrc + (kk + hi) * BSTRIDE + lo;
      const float* bp1 = bsrc + (kk + 2 + hi) * BSTRIDE + lo;
      v2f bb;
      bb.x = bp0[0];  bb.y = bp1[0];  acc0 = WMMA_F32(a, bb, acc0);
      bb.x = bp0[16]; bb.y = bp1[16]; acc1 = WMMA_F32(a, bb, acc1);
      bb.x = bp0[32]; bb.y = bp1[32]; acc2 = WMMA_F32(a, bb, acc2);
      bb.x = bp0[48]; bb.y = bp1[48]; acc3 = WMMA_F32(a, bb, acc3);
    }
  }

  const int orow = rowbase + wave * 16 + hi * 8;   // C/D: lanes 0-15 -> M=v, 16-31 -> M=v+8
  v8f accs[4] = {acc0, acc1, acc2, acc3};
  for (int t = 0; t < 4; ++t) {
    const int col = j0 + t * 16 + lo;
    const float* ap = (const float*)&accs[t];
    for (int v = 0; v < 8; ++v) {
      const int o = orow + v;
      H[(size_t)o * NCOL + col] = ap[v] + bias0[o];
    }
  }
}

// ---------------------------------------------------------------------------
// Kernel 3: per-channel batch stats -> folded BN scale/shift.
// ---------------------------------------------------------------------------
__global__ __launch_bounds__(256) void k_bnstats(
    const float* __restrict__ H, const float* __restrict__ gamma,
    const float* __restrict__ beta, float* __restrict__ bnscale,
    float* __restrict__ bnshift)
{
  __shared__ float rs[256], rq[256];
  const int o = blockIdx.x, tid = threadIdx.x;
  float s = 0.0f, q = 0.0f;
  const float* row = H + (size_t)o * NCOL;
  for (int c4 = tid * 4; c4 < NCOL; c4 += 1024) {
    v4f h = *(const v4f*)(row + c4);
    s += h.x + h.y + h.z + h.w;
    q += h.x * h.x + h.y * h.y + h.z * h.z + h.w * h.w;
  }
  rs[tid] = s; rq[tid] = q;
  __syncthreads();
  for (int off = 128; off > 0; off >>= 1) {
    if (tid < off) { rs[tid] += rs[tid + off]; rq[tid] += rq[tid + off]; }
    __syncthreads();
  }
  if (tid == 0) {
    const float inv = 1.0f / (float)NCOL;
    float mean = rs[0] * inv;
    float var  = rq[0] * inv - mean * mean;
    float sc   = gamma[o] * __frsqrt_rn(var + 1e-5f);
    bnscale[o] = sc;
    bnshift[o] = beta[o] - mean * sc;
  }
}

// ---------------------------------------------------------------------------
// Kernel 4: conv1 GEMM; BN+ReLU fused into B-operand (post-LDS-load),
// bias+ReLU epilogue, scatter to (b, o, n) output layout.
// ---------------------------------------------------------------------------
__global__ __launch_bounds__(256) void k_conv1_wmma(
    const float* __restrict__ H, const float* __restrict__ W1,
    const float* __restrict__ bias1, const float* __restrict__ bnscale,
    const float* __restrict__ bnshift, float* __restrict__ out)
{
  __shared__ float As[128 * ASTRIDE];
  __shared__ float Bs[2][KCH * BSTRIDE];
  __shared__ float ssc[CMID], ssh[CMID];
  const int tid  = threadIdx.x;
  const int lane = tid & 31;
  const int wave = tid >> 5;
  const int lo = lane & 15, hi = lane >> 4;
  const int rowbase = blockIdx.y * 128;
  const int j0 = blockIdx.x * 64;

  ssc[tid] = bnscale[tid];
  ssh[tid] = bnshift[tid];

  v8f acc0 = {}, acc1 = {}, acc2 = {}, acc3 = {};

  constexpr int nchunk = CMID / KCH;           // 4
  if (wave == 0) {
    tdm_load_tile_2d((unsigned)(uintptr_t)&Bs[0][0], H + j0,
                     NCOL, CMID, 64, KCH, NCOL);
  }

  for (int c = 0; c < nchunk; ++c) {
    const int kc = c * KCH;
    __syncthreads();
    for (int i = tid; i < 128 * 32; i += 256) {
      int m = i >> 5, k2 = (i & 31) << 1;
      *(v2f*)(As + m * ASTRIDE + k2) =
          *(const v2f*)(W1 + (size_t)(rowbase + m) * CMID + kc + k2);
    }
    if (wave == 0) {
      if (c + 1 < nchunk) {
        tdm_load_tile_2d((unsigned)(uintptr_t)&Bs[(c + 1) & 1][0],
                         H + (size_t)(kc + KCH) * NCOL + j0,
                         NCOL, (unsigned)(CMID - kc - KCH), 64, KCH, NCOL);
        __builtin_amdgcn_s_wait_tensorcnt(1);
      } else {
        __builtin_amdgcn_s_wait_tensorcnt(0);
      }
    }
    __syncthreads();

    const float* bsrc = Bs[c & 1];
    const float* arow = As + (wave * 16 + lo) * ASTRIDE + 2 * hi;
    for (int kk = 0; kk < KCH; kk += 4) {
      const v2f a = *(const v2f*)(arow + kk);
      const int kb0 = kc + kk + hi, kb1 = kc + kk + 2 + hi;
      const float sc0 = ssc[kb0], sh0 = ssh[kb0];
      const float sc1 = ssc[kb1], sh1 = ssh[kb1];
      const float* bp0 = bsrc + (kk + hi) * BSTRIDE + lo;
      const float* bp1 = bsrc + (kk + 2 + hi) * BSTRIDE + lo;
      v2f bb;
      bb.x = fmaxf(sc0 * bp0[0]  + sh0, 0.0f);
      bb.y = fmaxf(sc1 * bp1[0]  + sh1, 0.0f); acc0 = WMMA_F32(a, bb, acc0);
      bb.x = fmaxf(sc0 * bp0[16] + sh0, 0.0f);
      bb.y = fmaxf(sc1 * bp1[16] + sh1, 0.0f); acc1 = WMMA_F32(a, bb, acc1);
      bb.x = fmaxf(sc0 * bp0[32] + sh0, 0.0f);
      bb.y = fmaxf(sc1 * bp1[32] + sh1, 0.0f); acc2 = WMMA_F32(a, bb, acc2);
      bb.x = fmaxf(sc0 * bp0[48] + sh0, 0.0f);
      bb.y = fmaxf(sc1 * bp1[48] + sh1, 0.0f); acc3 = WMMA_F32(a, bb, acc3);
    }
  }

  const int orow = rowbase + wave * 16 + hi * 8;
  v8f accs[4] = {acc0, acc1, acc2, acc3};
  for (int t = 0; t < 4; ++t) {
    const int col = j0 + t * 16 + lo;
    const int bcol = col >> 13;          // col / N
    const int ncol = col & (NN - 1);     // col % N
    const float* ap = (const float*)&accs[t];
    for (int v = 0; v < 8; ++v) {
      const int o = orow + v;
      float r = ap[v] + bias1[o];
      out[(size_t)bcol * COUT * NN + (size_t)o * NN + ncol] = fmaxf(r, 0.0f);
    }
  }
}

// ---------------------------------------------------------------------------
extern "C" void kernel_launch(void* const* d_in, const int* in_sizes, int n_in,
                              void* d_out, int out_size, void* d_ws, size_t ws_size,
                              hipStream_t stream) {
  (void)in_sizes; (void)n_in; (void)out_size; (void)ws_size;
  const float* xyz1    = (const float*)d_in[0];
  const float* xyz2    = (const float*)d_in[1];
  const float* points1 = (const float*)d_in[2];
  const float* points2 = (const float*)d_in[3];
  const float* W0      = (const float*)d_in[4];
  const float* b0      = (const float*)d_in[5];
  const float* gamma0  = (const float*)d_in[6];
  const float* beta0   = (const float*)d_in[7];
  const float* W1      = (const float*)d_in[8];
  const float* b1      = (const float*)d_in[9];

  float* wsf = (float*)d_ws;
  float* NP      = wsf;                               // 384 x 65536 (96 MB)
  float* H       = NP + (size_t)CIN * NCOL;           // 256 x 65536 (64 MB)
  float* bnscale = H + (size_t)CMID * NCOL;           // 256
  float* bnshift = bnscale + CMID;                    // 256

  k_topk_interp<<<dim3(BB * (NN / 64)), dim3(256), 0, stream>>>(
      xyz1, xyz2, points1, points2, NP);
  k_conv0_wmma<<<dim3(NCOL / 64, 2), dim3(256), 0, stream>>>(NP, W0, b0, H);
  k_bnstats<<<dim3(CMID), dim3(256), 0, stream>>>(H, gamma0, beta0, bnscale, bnshift);
  k_conv1_wmma<<<dim3(NCOL / 64, 2), dim3(256), 0, stream>>>(
      H, W1, b1, bnscale, bnshift, (float*)d_out);
}